// GatLayer_60146722013383
// MI455X (gfx1250) — compile-verified
//
#include <hip/hip_runtime.h>
#include <math.h>

typedef __attribute__((ext_vector_type(2))) float v2f;
typedef __attribute__((ext_vector_type(8))) float v8f;

#define IN_DIM  128
#define OUT_DIM 64
#define LEAKY   0.2f

// ---------------------------------------------------------------------------
// K0: init segmax = -inf, segsum = 0, out = 0
// ---------------------------------------------------------------------------
__global__ void gat_init(float* __restrict__ segmax, float* __restrict__ segsum,
                         float* __restrict__ out, int N) {
    int tid = blockIdx.x * blockDim.x + threadIdx.x;
    if (tid < N) {
        segmax[tid] = -INFINITY;
        segsum[tid] = 0.0f;
    }
    if (tid < N * OUT_DIM) out[tid] = 0.0f;
}

// ---------------------------------------------------------------------------
// K1: H = X @ W using V_WMMA_F32_16X16X4_F32.
// One wave computes a 16x64 output strip (4 accumulators of 16x16).
// A (16x4 f32): lanes 0-15 hold M=0..15 with VGPR0=K0,VGPR1=K1;
//               lanes 16-31 hold K2,K3.  B mirrors with N across lanes.
// C/D (16x16 f32): VGPR r: lanes0-15 -> (M=r, N=lane); lanes16-31 -> (M=r+8).
// ---------------------------------------------------------------------------
__global__ void gat_gemm(const float* __restrict__ X, const float* __restrict__ W,
                         float* __restrict__ H, int N) {
    int wave = (blockIdx.x * blockDim.x + threadIdx.x) >> 5;
    int lane = threadIdx.x & 31;
    int row_base = wave * 16;
    if (row_base >= N) return;

    int half = lane >> 4;          // 0 or 1
    int sub  = lane & 15;          // row (A) / col (B) within tile
    int kgrp = half * 2;           // K offset handled by this half-wave
    int mrow = row_base + sub;

    v8f c0 = {}, c1 = {}, c2 = {}, c3 = {};
    const float* xrow = X + (size_t)mrow * IN_DIM;

    #pragma unroll 4
    for (int k = 0; k < IN_DIM; k += 4) {
        v2f a;
        a.x = xrow[k + kgrp + 0];
        a.y = xrow[k + kgrp + 1];
        const float* w0 = W + (size_t)(k + kgrp) * OUT_DIM;
        const float* w1 = W + (size_t)(k + kgrp + 1) * OUT_DIM;

        v2f b0, b1, b2, b3;
        b0.x = w0[sub +  0]; b0.y = w1[sub +  0];
        b1.x = w0[sub + 16]; b1.y = w1[sub + 16];
        b2.x = w0[sub + 32]; b2.y = w1[sub + 32];
        b3.x = w0[sub + 48]; b3.y = w1[sub + 48];

        c0 = __builtin_amdgcn_wmma_f32_16x16x4_f32(false, a, false, b0, (short)0, c0, false, false);
        c1 = __builtin_amdgcn_wmma_f32_16x16x4_f32(false, a, false, b1, (short)0, c1, false, false);
        c2 = __builtin_amdgcn_wmma_f32_16x16x4_f32(false, a, false, b2, (short)0, c2, false, false);
        c3 = __builtin_amdgcn_wmma_f32_16x16x4_f32(false, a, false, b3, (short)0, c3, false, false);
    }

    #pragma unroll
    for (int r = 0; r < 8; ++r) {
        size_t orow = (size_t)(row_base + r + half * 8) * OUT_DIM;
        H[orow + sub +  0] = c0[r];
        H[orow + sub + 16] = c1[r];
        H[orow + sub + 32] = c2[r];
        H[orow + sub + 48] = c3[r];
    }
}

// ---------------------------------------------------------------------------
// K2: per-node attention scalars  s[i] = h[i].a_src,  d[i] = h[i].a_dst
// One wave (32 lanes) per node; float2 per lane; shfl-xor reduction.
// ---------------------------------------------------------------------------
__global__ void gat_sd(const float* __restrict__ H, const float* __restrict__ bias,
                       float* __restrict__ s, float* __restrict__ dv, int N) {
    int wave = (blockIdx.x * blockDim.x + threadIdx.x) >> 5;
    int lane = threadIdx.x & 31;
    if (wave >= N) return;

    float h0 = H[(size_t)wave * OUT_DIM + lane * 2 + 0];
    float h1 = H[(size_t)wave * OUT_DIM + lane * 2 + 1];
    float ss = h0 * bias[lane * 2] + h1 * bias[lane * 2 + 1];
    float dd = h0 * bias[OUT_DIM + lane * 2] + h1 * bias[OUT_DIM + lane * 2 + 1];

    #pragma unroll
    for (int off = 16; off > 0; off >>= 1) {
        ss += __shfl_xor(ss, off, 32);
        dd += __shfl_xor(dd, off, 32);
    }
    if (lane == 0) { s[wave] = ss; dv[wave] = dd; }
}

// ---------------------------------------------------------------------------
// Edge helpers
// ---------------------------------------------------------------------------
__device__ __forceinline__ void edge_nodes(const long long* __restrict__ adj,
                                           int e, int E, int& src, int& dst) {
    if (e < E) { src = (int)adj[e]; dst = (int)adj[(size_t)E + e]; }
    else       { src = dst = e - E; }   // self-loop
}

__device__ __forceinline__ float leaky_logit(float t) {
    return t >= 0.0f ? t : LEAKY * t;
}

// signed-max for v>=0, unsigned-min for v<0 (init -inf = 0xFF800000)
__device__ __forceinline__ void atomic_max_float(float* addr, float v) {
    if (v >= 0.0f) atomicMax((int*)addr, __float_as_int(v));
    else           atomicMin((unsigned int*)addr, __float_as_uint(v));
}

// ---------------------------------------------------------------------------
// K3: segment max of leaky-relu logits keyed by src
// ---------------------------------------------------------------------------
__global__ void gat_edge_max(const long long* __restrict__ adj,
                             const float* __restrict__ s, const float* __restrict__ dv,
                             float* __restrict__ segmax, int E, int Et) {
    int e = blockIdx.x * blockDim.x + threadIdx.x;
    if (e >= Et) return;
    int src, dst;
    edge_nodes(adj, e, E, src, dst);
    atomic_max_float(&segmax[src], leaky_logit(s[src] + dv[dst]));
}

// ---------------------------------------------------------------------------
// K4: segment sum of exp(logit - segmax[src])
// ---------------------------------------------------------------------------
__global__ void gat_edge_sum(const long long* __restrict__ adj,
                             const float* __restrict__ s, const float* __restrict__ dv,
                             const float* __restrict__ segmax, float* __restrict__ segsum,
                             int E, int Et) {
    int e = blockIdx.x * blockDim.x + threadIdx.x;
    if (e >= Et) return;
    int src, dst;
    edge_nodes(adj, e, E, src, dst);
    float t = leaky_logit(s[src] + dv[dst]);
    atomicAdd(&segsum[src], __expf(t - segmax[src]));
}

// ---------------------------------------------------------------------------
// K5: out[src] += alpha * h[dst]    (16 lanes per edge, 4 floats per lane)
// ---------------------------------------------------------------------------
__global__ void gat_edge_accum(const long long* __restrict__ adj,
                               const float* __restrict__ s, const float* __restrict__ dv,
                               const float* __restrict__ segmax, const float* __restrict__ segsum,
                               const float* __restrict__ H, float* __restrict__ out,
                               int E, int Et) {
    int tid  = blockIdx.x * blockDim.x + threadIdx.x;
    int e    = tid >> 4;
    int part = tid & 15;
    if (e >= Et) return;
    int src, dst;
    edge_nodes(adj, e, E, src, dst);
    float t = leaky_logit(s[src] + dv[dst]);
    float alpha = __expf(t - segmax[src]) / segsum[src];

    const float4 hv = *(const float4*)(H + (size_t)dst * OUT_DIM + part * 4);
    float* o = out + (size_t)src * OUT_DIM + part * 4;
    atomicAdd(o + 0, alpha * hv.x);
    atomicAdd(o + 1, alpha * hv.y);
    atomicAdd(o + 2, alpha * hv.z);
    atomicAdd(o + 3, alpha * hv.w);
}

// ---------------------------------------------------------------------------
// Host-side launcher
// ---------------------------------------------------------------------------
extern "C" void kernel_launch(void* const* d_in, const int* in_sizes, int n_in,
                              void* d_out, int out_size, void* d_ws, size_t ws_size,
                              hipStream_t stream) {
    const float*     X    = (const float*)d_in[0];
    const long long* adj  = (const long long*)d_in[1];   // int64 (2,E)
    const float*     W    = (const float*)d_in[2];
    const float*     bias = (const float*)d_in[3];
    float*           out  = (float*)d_out;

    int N  = in_sizes[0] / IN_DIM;     // 100000
    int E  = in_sizes[1] / 2;          // 1600000
    int Et = E + N;                    // edges + self loops

    // workspace layout (floats): H[N*64] | s[N] | d[N] | segmax[N] | segsum[N]
    float* H      = (float*)d_ws;
    float* s      = H + (size_t)N * OUT_DIM;
    float* dv     = s + N;
    float* segmax = dv + N;
    float* segsum = segmax + N;

    const int BLK = 256;

    // K0: init
    {
        int total = N * OUT_DIM;
        gat_init<<<(total + BLK - 1) / BLK, BLK, 0, stream>>>(segmax, segsum, out, N);
    }
    // K1: WMMA GEMM (one wave per 16 rows)
    {
        int waves   = (N + 15) / 16;
        int threads = waves * 32;
        gat_gemm<<<(threads + BLK - 1) / BLK, BLK, 0, stream>>>(X, W, H, N);
    }
    // K2: per-node scalars (one wave per node)
    {
        long long threads = (long long)N * 32;
        gat_sd<<<(int)((threads + BLK - 1) / BLK), BLK, 0, stream>>>(H, bias, s, dv, N);
    }
    // K3: segment max
    gat_edge_max<<<(Et + BLK - 1) / BLK, BLK, 0, stream>>>(adj, s, dv, segmax, E, Et);
    // K4: segment sum
    gat_edge_sum<<<(Et + BLK - 1) / BLK, BLK, 0, stream>>>(adj, s, dv, segmax, segsum, E, Et);
    // K5: weighted scatter-add
    {
        long long threads = (long long)Et * 16;
        gat_edge_accum<<<(int)((threads + BLK - 1) / BLK), BLK, 0, stream>>>(
            adj, s, dv, segmax, segsum, H, out, E, Et);
    }
}